// GCN_76287209112018
// MI455X (gfx1250) — compile-verified
//
#include <hip/hip_runtime.h>
#include <stdint.h>

#define N_NODES 100000
#define N_EDGES 1000000
#define HDIM    64
#define NPAD    100096   // 782 * 128, multiple of 16-row WMMA tiles * 8 waves
#define NGRAPH  64

typedef float v2f __attribute__((ext_vector_type(2)));
typedef float v8f __attribute__((ext_vector_type(8)));

// ---------------------------------------------------------------- zero fill
__global__ void gcn_zero_kernel(float* __restrict__ p, int n) {
  int i = blockIdx.x * blockDim.x + threadIdx.x;
  if (i < n) p[i] = 0.0f;
}

// ------------------------------------------------- edge gather + scatter-add
// agg[dst] += feat[src]; one thread = (edge, 4-feature chunk)
__global__ void gcn_scatter_kernel(const float* __restrict__ feat,
                                   const int* __restrict__ src,
                                   const int* __restrict__ dst,
                                   float* __restrict__ agg) {
  int i = blockIdx.x * blockDim.x + threadIdx.x;
  int e = i >> 4;
  int c = (i & 15) << 2;
  if (e >= N_EDGES) return;
  int s = src[e];
  int d = dst[e];
  const float4 v = *(const float4*)(feat + (size_t)s * HDIM + c);
  float* o = agg + (size_t)d * HDIM + c;
  atomicAdd(o + 0, v.x);
  atomicAdd(o + 1, v.y);
  atomicAdd(o + 2, v.z);
  atomicAdd(o + 3, v.w);
}

// CDNA5 direct memory->LDS async copy (16B per lane), tracked by ASYNCcnt.
// lds_off = wave-relative LDS byte offset (low 32 bits of flat shared addr).
__device__ __forceinline__ void async_copy_b128(uint32_t lds_off,
                                                const void* gaddr) {
  asm volatile("global_load_async_to_lds_b128 %0, %1, off"
               :: "v"(lds_off), "v"((uint64_t)(uintptr_t)gaddr)
               : "memory");
}

// --------------------------------------------------- fused GraphConv + PReLU
// out = prelu(agg @ Wr + xin @ Wo + bias)
// One wave computes a 16x64 tile via V_WMMA_F32_16X16X4_F32 (fp32 exact).
// agg/out are NPAD-padded (pad rows zero / don't-care); xin reads guarded.
__global__ __launch_bounds__(256) void gcn_conv_wmma_kernel(
    const float* __restrict__ agg,
    const float* __restrict__ xin,
    const float* __restrict__ Wr,
    const float* __restrict__ Wo,
    const float* __restrict__ bias,
    const float* __restrict__ slope_ptr, int use_prelu,
    float* __restrict__ out) {
  __shared__ float sWr[HDIM * HDIM];
  __shared__ float sWo[HDIM * HDIM];

  // Stage both 16KB weight matrices into LDS via GLOBAL_LOAD_ASYNC_TO_LDS_B128
  // (no VGPR round-trip): 256 threads * 4 iters * 16B = 16KB per matrix.
#pragma unroll
  for (int it = 0; it < 4; ++it) {
    const int idx = (threadIdx.x + it * 256) * 4;   // float index, 16B chunk
    async_copy_b128((uint32_t)(uintptr_t)(sWr + idx), Wr + idx);
    async_copy_b128((uint32_t)(uintptr_t)(sWo + idx), Wo + idx);
  }
  asm volatile("s_wait_asynccnt 0x0" ::: "memory");
  __syncthreads();

  const int lane = threadIdx.x & 31;
  const int wave = threadIdx.x >> 5;
  const int half = lane >> 4;   // 0: K base +0 ; 1: K base +2 (A/B frag halves)
  const int m    = lane & 15;
  const int r0   = (blockIdx.x * 8 + wave) * 16;   // 16 rows per wave

  // init accumulators with bias: D element (M=j+8*half, N=t*16+m)
  v8f acc[4];
#pragma unroll
  for (int t = 0; t < 4; ++t) {
    float bv = bias[t * 16 + m];
#pragma unroll
    for (int j = 0; j < 8; ++j) acc[t][j] = bv;
  }

  const int rowA = r0 + m;
  const float* aggp = agg + (size_t)rowA * HDIM + 2 * half;
  const float* xp   = xin + (size_t)rowA * HDIM + 2 * half;
  const bool xok = (rowA < N_NODES);

#pragma unroll
  for (int k = 0; k < HDIM; k += 4) {
    // A fragments (16x4 f32): lane m holds row m, VGPR0/1 = K=kk, kk+1
    v2f aA = *(const v2f*)(aggp + k);
    v2f aX = {0.0f, 0.0f};
    if (xok) aX = *(const v2f*)(xp + k);   // reconverges before WMMA
    const int kk = k + 2 * half;
#pragma unroll
    for (int t = 0; t < 4; ++t) {
      const int c = t * 16 + m;
      v2f bR, bO;                          // B fragments (4x16) from LDS
      bR.x = sWr[kk * HDIM + c];
      bR.y = sWr[(kk + 1) * HDIM + c];
      bO.x = sWo[kk * HDIM + c];
      bO.y = sWo[(kk + 1) * HDIM + c];
      acc[t] = __builtin_amdgcn_wmma_f32_16x16x4_f32(
          false, aA, false, bR, (short)0, acc[t], false, false);
      acc[t] = __builtin_amdgcn_wmma_f32_16x16x4_f32(
          false, aX, false, bO, (short)0, acc[t], false, false);
    }
  }

  float slope = 1.0f;
  if (use_prelu) slope = *slope_ptr;
#pragma unroll
  for (int t = 0; t < 4; ++t) {
    const int c = t * 16 + m;
#pragma unroll
    for (int j = 0; j < 8; ++j) {
      const int row = r0 + half * 8 + j;
      float v = acc[t][j];
      if (use_prelu && v < 0.0f) v *= slope;
      out[(size_t)row * HDIM + c] = v;
    }
  }
}

// ------------------------------------------------------- global mean pooling
__global__ void gcn_pool_kernel(const float* __restrict__ h,
                                const int* __restrict__ batch,
                                float* __restrict__ psum,
                                float* __restrict__ pcnt) {
  int i = blockIdx.x * blockDim.x + threadIdx.x;
  int n = i >> 4;
  int c = (i & 15) << 2;
  if (n >= N_NODES) return;
  int g = batch[n];
  const float4 v = *(const float4*)(h + (size_t)n * HDIM + c);
  float* o = psum + (size_t)g * HDIM + c;
  atomicAdd(o + 0, v.x);
  atomicAdd(o + 1, v.y);
  atomicAdd(o + 2, v.z);
  atomicAdd(o + 3, v.w);
  if (c == 0) atomicAdd(pcnt + g, 1.0f);
}

// ----------------------------------------------------------------- MLP head
__global__ void gcn_head_kernel(const float* __restrict__ psum,
                                const float* __restrict__ pcnt,
                                const float* __restrict__ W0,
                                const float* __restrict__ b0,
                                const float* __restrict__ W1,
                                const float* __restrict__ b1,
                                float* __restrict__ out) {
  int g = threadIdx.x;
  if (g >= NGRAPH) return;
  float cnt = pcnt[g];
  if (cnt < 1.0f) cnt = 1.0f;
  float p[HDIM];
#pragma unroll
  for (int k = 0; k < HDIM; ++k) p[k] = psum[g * HDIM + k] / cnt;
  float o = b1[0];
  for (int j = 0; j < HDIM; ++j) {
    float t = b0[j];
#pragma unroll
    for (int k = 0; k < HDIM; ++k) t += p[k] * W0[k * HDIM + j];
    o += t * W1[j];
  }
  out[g] = o;
}

// ---------------------------------------------------------------- dispatch
extern "C" void kernel_launch(void* const* d_in, const int* in_sizes, int n_in,
                              void* d_out, int out_size, void* d_ws, size_t ws_size,
                              hipStream_t stream) {
  const float* x     = (const float*)d_in[0];
  const int*   ei    = (const int*)d_in[1];
  const int*   batch = (const int*)d_in[2];
  const float* W1r = (const float*)d_in[3];
  const float* b1  = (const float*)d_in[4];
  const float* W1o = (const float*)d_in[5];
  const float* W2r = (const float*)d_in[6];
  const float* b2  = (const float*)d_in[7];
  const float* W2o = (const float*)d_in[8];
  const float* W3r = (const float*)d_in[9];
  const float* b3  = (const float*)d_in[10];
  const float* W3o = (const float*)d_in[11];
  const float* a1  = (const float*)d_in[12];
  const float* a2  = (const float*)d_in[13];
  const float* Wl0 = (const float*)d_in[14];
  const float* bl0 = (const float*)d_in[15];
  const float* Wl1 = (const float*)d_in[16];
  const float* bl1 = (const float*)d_in[17];
  const int* src = ei;
  const int* dst = ei + N_EDGES;

  float* ws   = (float*)d_ws;
  float* agg  = ws;                              // NPAD*64
  float* hA   = ws + (size_t)NPAD * HDIM;        // NPAD*64
  float* hB   = ws + (size_t)2 * NPAD * HDIM;    // NPAD*64
  float* psum = ws + (size_t)3 * NPAD * HDIM;    // G*64
  float* pcnt = psum + NGRAPH * HDIM;            // G

  const int aggN   = NPAD * HDIM;
  const int zeroB  = (aggN + 255) / 256;
  const int scatB  = (N_EDGES * 16 + 255) / 256;
  const int convB  = NPAD / 128;                 // 8 waves * 16 rows per block
  const int poolB  = (N_NODES * 16 + 255) / 256;
  const int pzN    = NGRAPH * HDIM + NGRAPH;
  const int pzB    = (pzN + 255) / 256;

  // ---- layer 1: agg = scatter(x); hA = prelu(agg@W1r + x@W1o + b1)
  gcn_zero_kernel<<<zeroB, 256, 0, stream>>>(agg, aggN);
  gcn_scatter_kernel<<<scatB, 256, 0, stream>>>(x, src, dst, agg);
  gcn_conv_wmma_kernel<<<convB, 256, 0, stream>>>(agg, x, W1r, W1o, b1, a1, 1, hA);

  // ---- layer 2
  gcn_zero_kernel<<<zeroB, 256, 0, stream>>>(agg, aggN);
  gcn_scatter_kernel<<<scatB, 256, 0, stream>>>(hA, src, dst, agg);
  gcn_conv_wmma_kernel<<<convB, 256, 0, stream>>>(agg, hA, W2r, W2o, b2, a2, 1, hB);

  // ---- layer 3 (no activation)
  gcn_zero_kernel<<<zeroB, 256, 0, stream>>>(agg, aggN);
  gcn_scatter_kernel<<<scatB, 256, 0, stream>>>(hB, src, dst, agg);
  gcn_conv_wmma_kernel<<<convB, 256, 0, stream>>>(agg, hB, W3r, W3o, b3, nullptr, 0, hA);

  // ---- mean pool + head
  gcn_zero_kernel<<<pzB, 256, 0, stream>>>(psum, pzN);
  gcn_pool_kernel<<<poolB, 256, 0, stream>>>(hA, batch, psum, pcnt);
  gcn_head_kernel<<<1, 64, 0, stream>>>(psum, pcnt, Wl0, bl0, Wl1, bl1, (float*)d_out);
}